// GroupWiseAgg_48558900249222
// MI455X (gfx1250) — compile-verified
//
#include <hip/hip_runtime.h>
#include <math.h>

#define GROUPS 8
#define NVIEW  4
#define C_     32
#define D_     48
#define H_     112
#define W_     144
#define HW     (H_*W_)

typedef __attribute__((ext_vector_type(2))) float v2f;
typedef __attribute__((ext_vector_type(8))) float v8f;

// ---------------------------------------------------------------------------
// Setup: compute per-view warp matrices  M = fold(src) * inv(fold(ref))
// ---------------------------------------------------------------------------
__device__ inline void mat4_mul(const float A[16], const float B[16], float Cm[16]) {
  for (int r = 0; r < 4; ++r)
    for (int c = 0; c < 4; ++c) {
      float s = 0.f;
      for (int k = 0; k < 4; ++k) s += A[r*4+k] * B[k*4+c];
      Cm[r*4+c] = s;
    }
}

__device__ inline void mat4_inv(const float m[16], float o[16]) {
  float inv[16];
  inv[0]  =  m[5]*m[10]*m[15] - m[5]*m[11]*m[14] - m[9]*m[6]*m[15] + m[9]*m[7]*m[14] + m[13]*m[6]*m[11] - m[13]*m[7]*m[10];
  inv[4]  = -m[4]*m[10]*m[15] + m[4]*m[11]*m[14] + m[8]*m[6]*m[15] - m[8]*m[7]*m[14] - m[12]*m[6]*m[11] + m[12]*m[7]*m[10];
  inv[8]  =  m[4]*m[9]*m[15]  - m[4]*m[11]*m[13] - m[8]*m[5]*m[15] + m[8]*m[7]*m[13] + m[12]*m[5]*m[11] - m[12]*m[7]*m[9];
  inv[12] = -m[4]*m[9]*m[14]  + m[4]*m[10]*m[13] + m[8]*m[5]*m[14] - m[8]*m[6]*m[13] - m[12]*m[5]*m[10] + m[12]*m[6]*m[9];
  inv[1]  = -m[1]*m[10]*m[15] + m[1]*m[11]*m[14] + m[9]*m[2]*m[15] - m[9]*m[3]*m[14] - m[13]*m[2]*m[11] + m[13]*m[3]*m[10];
  inv[5]  =  m[0]*m[10]*m[15] - m[0]*m[11]*m[14] - m[8]*m[2]*m[15] + m[8]*m[3]*m[14] + m[12]*m[2]*m[11] - m[12]*m[3]*m[10];
  inv[9]  = -m[0]*m[9]*m[15]  + m[0]*m[11]*m[13] + m[8]*m[1]*m[15] - m[8]*m[3]*m[13] - m[12]*m[1]*m[11] + m[12]*m[3]*m[9];
  inv[13] =  m[0]*m[9]*m[14]  - m[0]*m[10]*m[13] - m[8]*m[1]*m[14] + m[8]*m[2]*m[13] + m[12]*m[1]*m[10] - m[12]*m[2]*m[9];
  inv[2]  =  m[1]*m[6]*m[15]  - m[1]*m[7]*m[14]  - m[5]*m[2]*m[15] + m[5]*m[3]*m[14] + m[13]*m[2]*m[7]  - m[13]*m[3]*m[6];
  inv[6]  = -m[0]*m[6]*m[15]  + m[0]*m[7]*m[14]  + m[4]*m[2]*m[15] - m[4]*m[3]*m[14] - m[12]*m[2]*m[7]  + m[12]*m[3]*m[6];
  inv[10] =  m[0]*m[5]*m[15]  - m[0]*m[7]*m[13]  - m[4]*m[1]*m[15] + m[4]*m[3]*m[13] + m[12]*m[1]*m[7]  - m[12]*m[3]*m[5];
  inv[14] = -m[0]*m[5]*m[14]  + m[0]*m[6]*m[13]  + m[4]*m[1]*m[14] - m[4]*m[2]*m[13] - m[12]*m[1]*m[6]  + m[12]*m[2]*m[5];
  inv[3]  = -m[1]*m[6]*m[11]  + m[1]*m[7]*m[10]  + m[5]*m[2]*m[11] - m[5]*m[3]*m[10] - m[9]*m[2]*m[7]   + m[9]*m[3]*m[6];
  inv[7]  =  m[0]*m[6]*m[11]  - m[0]*m[7]*m[10]  - m[4]*m[2]*m[11] + m[4]*m[3]*m[10] + m[8]*m[2]*m[7]   - m[8]*m[3]*m[6];
  inv[11] = -m[0]*m[5]*m[11]  + m[0]*m[7]*m[9]   + m[4]*m[1]*m[11] - m[4]*m[3]*m[9]  - m[8]*m[1]*m[7]   + m[8]*m[3]*m[5];
  inv[15] =  m[0]*m[5]*m[10]  - m[0]*m[6]*m[9]   - m[4]*m[1]*m[10] + m[4]*m[2]*m[9]  + m[8]*m[1]*m[6]   - m[8]*m[2]*m[5];
  float det = m[0]*inv[0] + m[1]*inv[4] + m[2]*inv[8] + m[3]*inv[12];
  det = 1.0f / det;
  for (int i = 0; i < 16; ++i) o[i] = inv[i] * det;
}

// fold: top 3x4 = K[:3,:3] @ E[:3,:4]; bottom row = E row 3
__device__ inline void fold_proj(const float* pm, int v, float P[16]) {
  const float* E = pm + (v*2 + 0)*16;
  const float* K = pm + (v*2 + 1)*16;
  for (int r = 0; r < 3; ++r)
    for (int c = 0; c < 4; ++c) {
      float s = 0.f;
      for (int k = 0; k < 3; ++k) s += K[r*4+k] * E[k*4+c];
      P[r*4+c] = s;
    }
  for (int c = 0; c < 4; ++c) P[12+c] = E[12+c];
}

__global__ void gwc_setup_kernel(const float* __restrict__ pm, float* __restrict__ mats) {
  if (threadIdx.x != 0 || blockIdx.x != 0) return;
  float Pref[16], Pinv[16];
  fold_proj(pm, 0, Pref);
  mat4_inv(Pref, Pinv);
  for (int v = 1; v < NVIEW; ++v) {
    float Ps[16], M[16];
    fold_proj(pm, v, Ps);
    mat4_mul(Ps, Pinv, M);
    float* o = mats + (v-1)*12;
    o[0] = M[0];  o[1] = M[1];  o[2]  = M[2];
    o[3] = M[4];  o[4] = M[5];  o[5]  = M[6];
    o[6] = M[8];  o[7] = M[9];  o[8]  = M[10];
    o[9] = M[3];  o[10] = M[7]; o[11] = M[11];
  }
}

// ---------------------------------------------------------------------------
// NCHW -> NHWC feature transpose (makes every bilinear tap one 128B line)
// ---------------------------------------------------------------------------
__global__ void gwc_nchw_to_nhwc(const float* __restrict__ feat, float* __restrict__ out) {
  int idx = blockIdx.x * blockDim.x + threadIdx.x;   // over NVIEW*HW pixels
  if (idx >= NVIEW*HW) return;
  int v = idx / HW, p = idx % HW;
  const float* src = feat + (size_t)v*C_*HW + p;
  float* dst = out + (size_t)idx*C_;
#pragma unroll
  for (int c = 0; c < C_; ++c) dst[c] = src[(size_t)c*HW];
}

// ---------------------------------------------------------------------------
// Main kernel: warp + bilinear accumulate + group correlation via WMMA f32.
// wave32: 16 positions per wave; lanes 0-15 supply K-channels {0,1} of each
// group, lanes 16-31 supply {2,3} (16x4 f32 A-matrix layout) -> the two lane
// halves gather disjoint channel pairs, no redundant sampling.
// Diagonal of D = A(16x4)*B(4x16) is the 16 per-position group correlations.
// ---------------------------------------------------------------------------
__global__ __launch_bounds__(256)
void gwc_volume_kernel(const float* __restrict__ featNHWC,
                       const float* __restrict__ depth,
                       const float* __restrict__ mats,
                       float* __restrict__ out,
                       int depthsPerBlock) {
  const int lane    = threadIdx.x & 31;
  const int wave    = threadIdx.x >> 5;
  const int tile    = blockIdx.x * 8 + wave;     // 1008 tiles of 16 positions
  const int posBase = tile * 16;
  const int p       = posBase + (lane & 15);
  const float x = (float)(p % W_);
  const float y = (float)(p / W_);

  // per-view rotated ray + translation (depth-independent)
  float RX[3][3], T[3][3];
#pragma unroll
  for (int v = 0; v < 3; ++v) {
    const float* m = mats + v*12;
    RX[v][0] = m[0]*x + m[1]*y + m[2];
    RX[v][1] = m[3]*x + m[4]*y + m[5];
    RX[v][2] = m[6]*x + m[7]*y + m[8];
    T[v][0] = m[9]; T[v][1] = m[10]; T[v][2] = m[11];
  }

  const bool hiHalf = (lane >= 16);
  const int  aoff = hiHalf ? 2 : 0;   // A-matrix K channel pair offset
  const int  roff = hiHalf ? 1 : 0;   // B-matrix K row pair offset (rows {0,2} / {1,3})

  // reference-view features for this position: registers for all depths
  const float* refBase = featNHWC + (size_t)p * C_;
  float rb0[GROUPS], rb1[GROUPS];
#pragma unroll
  for (int g = 0; g < GROUPS; ++g) {
    rb0[g] = refBase[g*4 + roff];
    rb1[g] = refBase[g*4 + roff + 2];
  }

  // store-side lane mapping (hoisted): diagonal owners are lanes 0-7 (M=0..7)
  // and lanes 24-31 (M=8..15)
  const bool hasDiag = (lane < 8) || (lane >= 24);
  const int  ppStore = posBase + ((lane < 8) ? lane : (lane - 16));

  const int dBase = blockIdx.y * depthsPerBlock;
  for (int dd = 0; dd < depthsPerBlock; ++dd) {
    const int d = dBase + dd;
    const float dep = depth[(size_t)d*HW + p];

    float accx[GROUPS], accy[GROUPS];
#pragma unroll
    for (int g = 0; g < GROUPS; ++g) { accx[g] = 0.f; accy[g] = 0.f; }

#pragma unroll
    for (int v = 0; v < 3; ++v) {
      float X = RX[v][0]*dep + T[v][0];
      float Y = RX[v][1]*dep + T[v][1];
      float Z = RX[v][2]*dep + T[v][2];
      float iz = 1.0f / Z;
      float px = X*iz, py = Y*iz;
      float fx = floorf(px), fy = floorf(py);
      float dx = px - fx,  dy = py - fy;
      int   x0 = (int)fx,  y0 = (int)fy;
      const float* vb = featNHWC + (size_t)(v+1)*HW*C_;
#pragma unroll
      for (int t = 0; t < 4; ++t) {
        int xi = x0 + (t & 1);
        int yi = y0 + (t >> 1);
        float wgt = ((t & 1) ? dx : 1.f - dx) * ((t >> 1) ? dy : 1.f - dy);
        bool valid = (xi >= 0) && (xi < W_) && (yi >= 0) && (yi < H_);
        wgt = valid ? wgt : 0.f;
        int xc = min(max(xi, 0), W_-1);
        int yc = min(max(yi, 0), H_-1);
        const float* sp = vb + (size_t)(yc*W_ + xc)*C_ + aoff;
#pragma unroll
        for (int g = 0; g < GROUPS; ++g) {
          accx[g] += wgt * sp[g*4 + 0];   // contiguous pair -> b64 load
          accy[g] += wgt * sp[g*4 + 1];
        }
      }
    }

    // 8 WMMAs (EXEC all-1s here), diagonal extracted unmasked into scalars
    float vals[GROUPS];
#pragma unroll
    for (int g = 0; g < GROUPS; ++g) {
      v2f A; A.x = accx[g]; A.y = accy[g];
      v2f B; B.x = rb0[g];  B.y = rb1[g];
      v8f Cd = {};
      Cd = __builtin_amdgcn_wmma_f32_16x16x4_f32(
          /*neg_a=*/false, A, /*neg_b=*/false, B,
          /*c_mod=*/(short)0, Cd, /*reuse_a=*/false, /*reuse_b=*/false);
      float val = Cd[0];
#pragma unroll
      for (int r = 1; r < 8; ++r)
        if ((lane & 7) == r) val = Cd[r];
      vals[g] = val * (1.0f/12.0f);       // mean over 4 chans, / (nview-1)
    }

    // single masked store region per depth (one saveexec instead of eight)
    if (hasDiag) {
      float* ob = out + (size_t)d*HW + ppStore;
#pragma unroll
      for (int g = 0; g < GROUPS; ++g)
        ob[(size_t)g*D_*HW] = vals[g];
    }
  }
}

// ---------------------------------------------------------------------------
extern "C" void kernel_launch(void* const* d_in, const int* in_sizes, int n_in,
                              void* d_out, int out_size, void* d_ws, size_t ws_size,
                              hipStream_t stream) {
  const float* features = (const float*)d_in[0];   // (4,1,32,112,144)
  const float* projs    = (const float*)d_in[1];   // (1,4,2,4,4)
  const float* depth    = (const float*)d_in[2];   // (1,48,112,144)
  float* out = (float*)d_out;

  float* mats     = (float*)d_ws;                  // 36 floats
  float* featNHWC = (float*)d_ws + 64;             // 4*HW*32 floats (~8.3 MB)

  gwc_setup_kernel<<<1, 32, 0, stream>>>(projs, mats);

  int npix = NVIEW * HW;
  gwc_nchw_to_nhwc<<<(npix + 255)/256, 256, 0, stream>>>(features, featNHWC);

  // 1008 position tiles / 8 waves per block = 126 blocks; 8 depth slabs of 6
  dim3 grid(126, 8);
  gwc_volume_kernel<<<grid, 256, 0, stream>>>(featNHWC, depth, mats, out, 6);
}